// Layer_26774826123945
// MI455X (gfx1250) — compile-verified
//
#include <hip/hip_runtime.h>
#include <stdint.h>

// Projection: min ||w-mu||^2 s.t. |sum w| <= CAP, -BOX <= w <= BOX  (per row)
#define BOX    0.5f
#define CAP    0.1f
#define BITERS 60
#define NCOL   128
#define WPB    8              // waves per block
#define TPB    (WPB * 32)

typedef int v4i __attribute__((ext_vector_type(4)));
typedef __attribute__((address_space(1))) v4i as1_v4i;   // global
typedef __attribute__((address_space(3))) v4i as3_v4i;   // LDS
typedef __attribute__((address_space(3))) void as3_void;

// ---- CDNA5 async global->LDS copy (ASYNCcnt-tracked), 16B per lane ----
__device__ __forceinline__ void async_row_b128(const float* g, float* l) {
#if __has_builtin(__builtin_amdgcn_global_load_async_to_lds_b128)
  __builtin_amdgcn_global_load_async_to_lds_b128(
      (as1_v4i*)(void*)g, (as3_v4i*)(void*)l, 0, 0);
#else
  unsigned int       loff = (unsigned int)(uintptr_t)(as3_void*)(void*)l;
  unsigned long long ga   = (unsigned long long)(uintptr_t)g;
  asm volatile("global_load_async_to_lds_b128 %0, %1, off"
               :: "v"(loff), "v"(ga) : "memory");
#endif
}

template <int N>
__device__ __forceinline__ void wait_async() {
#if __has_builtin(__builtin_amdgcn_s_wait_asynccnt)
  __builtin_amdgcn_s_wait_asynccnt(N);
#else
  asm volatile("s_wait_asynccnt %0" :: "i"(N) : "memory");
#endif
}

// ---- wave32 butterfly reductions: result replicated to all lanes ----
__device__ __forceinline__ float wred_sum(float v) {
#pragma unroll
  for (int m = 16; m; m >>= 1) v += __shfl_xor(v, m, 32);
  return v;
}
__device__ __forceinline__ float wred_min(float v) {
#pragma unroll
  for (int m = 16; m; m >>= 1) v = fminf(v, __shfl_xor(v, m, 32));
  return v;
}
__device__ __forceinline__ float wred_max(float v) {
#pragma unroll
  for (int m = 16; m; m >>= 1) v = fmaxf(v, __shfl_xor(v, m, 32));
  return v;
}
__device__ __forceinline__ float clipb(float x) {
  return fminf(fmaxf(x, -BOX), BOX);
}

__global__ __launch_bounds__(TPB)
void proj_rows(const float* __restrict__ x, float* __restrict__ out,
               int rows, int wstride) {
  // per-wave double buffer: 8 waves * 2 * 512B = 8 KB LDS / block
  __shared__ __align__(16) float stage[WPB][2][NCOL];

  const int lane = threadIdx.x & 31;
  const int wv   = threadIdx.x >> 5;
  const int col  = lane << 2;                       // 4 floats per lane
  long long row  = (long long)blockIdx.x * WPB + wv;

  if (row < rows)
    async_row_b128(x + row * NCOL + col, &stage[wv][0][col]);

  int pb = 0;
  for (; row < rows; row += wstride) {
    const long long nrow = row + wstride;
    const bool has_next = (nrow < rows);
    if (has_next)                                    // prefetch next row
      async_row_b128(x + nrow * NCOL + col, &stage[wv][pb ^ 1][col]);
    // async loads complete in order: <=1 outstanding => current row landed
    if (has_next) wait_async<1>(); else wait_async<0>();

    const float4 mu = *(const float4*)&stage[wv][pb][col];
    const float m0 = mu.x, m1 = mu.y, m2 = mu.z, m3 = mu.w;

    // t = clip(sum(clip(mu)), -CAP, CAP)
    const float s0 = wred_sum(clipb(m0) + clipb(m1) + clipb(m2) + clipb(m3));
    const float t  = fminf(fmaxf(s0, -CAP), CAP);

    // bisection bracket
    float lo = wred_min(fminf(fminf(m0, m1), fminf(m2, m3))) - (BOX + CAP);
    float hi = wred_max(fmaxf(fmaxf(m0, m1), fmaxf(m2, m3))) + (BOX + CAP);

#pragma unroll 4
    for (int it = 0; it < BITERS; ++it) {
      const float mid = 0.5f * (lo + hi);
      float g = clipb(m0 - mid) + clipb(m1 - mid)
              + clipb(m2 - mid) + clipb(m3 - mid);
      g = wred_sum(g);                               // identical in all lanes
      const bool big = (g > t);
      lo = big ? mid : lo;
      hi = big ? hi : mid;
    }
    const float lam_star = 0.5f * (lo + hi);

    // closed-form lambda over the active set (exact, matches reference)
    const float z0 = m0 - lam_star, z1 = m1 - lam_star,
                z2 = m2 - lam_star, z3 = m3 - lam_star;
    const float a0 = (z0 > -BOX && z0 < BOX) ? 1.f : 0.f;
    const float a1 = (z1 > -BOX && z1 < BOX) ? 1.f : 0.f;
    const float a2 = (z2 > -BOX && z2 < BOX) ? 1.f : 0.f;
    const float a3 = (z3 > -BOX && z3 < BOX) ? 1.f : 0.f;
    const float h0 = ((z0 >= BOX) ? 1.f : 0.f) - ((z0 <= -BOX) ? 1.f : 0.f);
    const float h1 = ((z1 >= BOX) ? 1.f : 0.f) - ((z1 <= -BOX) ? 1.f : 0.f);
    const float h2 = ((z2 >= BOX) ? 1.f : 0.f) - ((z2 <= -BOX) ? 1.f : 0.f);
    const float h3 = ((z3 >= BOX) ? 1.f : 0.f) - ((z3 <= -BOX) ? 1.f : 0.f);

    const float n_act = wred_sum(a0 + a1 + a2 + a3);
    const float d_hl  = wred_sum(h0 + h1 + h2 + h3);          // n_hi - n_lo
    const float s_act = wred_sum(a0 * m0 + a1 * m1 + a2 * m2 + a3 * m3);

    float lam = (s_act + BOX * d_hl - t) / fmaxf(n_act, 1.f);
    lam = (n_act > 0.f) ? lam : lam_star;

    float4 w;
    w.x = clipb(m0 - lam); w.y = clipb(m1 - lam);
    w.z = clipb(m2 - lam); w.w = clipb(m3 - lam);
    *(float4*)(out + row * NCOL + col) = w;                   // global_store_b128

    pb ^= 1;
  }
}

extern "C" void kernel_launch(void* const* d_in, const int* in_sizes, int n_in,
                              void* d_out, int out_size, void* d_ws, size_t ws_size,
                              hipStream_t stream) {
  (void)n_in; (void)out_size; (void)d_ws; (void)ws_size;
  const float* x  = (const float*)d_in[0];
  float* out      = (float*)d_out;
  const int n     = in_sizes[0];
  const int rows  = n / NCOL;

  int blocks = (rows + WPB - 1) / WPB;
  if (blocks > 4096) blocks = 4096;          // ~8 rows per wave: pipeline depth
  const int wstride = blocks * WPB;          // grid-stride in units of waves

  proj_rows<<<blocks, TPB, 0, stream>>>(x, out, rows, wstride);
}